// GDMLTorchPredict_21431886807103
// MI455X (gfx1250) — compile-verified
//
#include <hip/hip_runtime.h>
#include <hip/hip_bf16.h>
#include <math.h>

// ---------------------------------------------------------------------------
// GDML force-field prediction, factorized into 4 fp32 WMMA GEMMs.
//   M=64 geometries, N=21 atoms, D=210 descriptors, T=12000 training rows.
//   Uses V_WMMA_F32_16X16X4_F32 (exact fp32 semantics, matches reference).
// ---------------------------------------------------------------------------

typedef __attribute__((ext_vector_type(2))) float v2f;
typedef __attribute__((ext_vector_type(8))) float v8f;

#define NATOMS 21
#define DESC   210      // 21*20/2
#define TROWS  12000
#define MBATCH 64
#define TTILES (TROWS / 16)   // 750 per-m partial slots

__device__ __forceinline__ constexpr float kQ()  { return 0.22360679774997896f; } // sqrt(5)/sig
__device__ __forceinline__ constexpr float kQ2() { return 0.05f;                } // q^2
__device__ __forceinline__ constexpr float kK0() { return 5.0f / 300.0f;        } // 5/(3*sig^2)

// ---------------------------------------------------------------------------
// k0: per-geometry descriptors. xs[m,p] = 1/dist(i_p,j_p); xs_n2[m] = q^2*||xs||^2
// ---------------------------------------------------------------------------
__global__ void k0_descr(const float* __restrict__ Rs,
                         float* __restrict__ xs, float* __restrict__ xs_n2) {
  __shared__ float sR[NATOMS * 3];
  __shared__ float sSum[8];
  const int m = blockIdx.x, tid = threadIdx.x;
  if (tid < NATOMS * 3) sR[tid] = Rs[m * NATOMS * 3 + tid];
  __syncthreads();
  float local = 0.0f;
  if (tid < DESC) {
    int p = tid;
    int i = 1;
    while (p >= (i * (i + 1)) / 2) ++i;      // tril_indices(N,-1) row-major inverse
    int j = p - (i * (i - 1)) / 2;
    float dx = sR[i * 3 + 0] - sR[j * 3 + 0];
    float dy = sR[i * 3 + 1] - sR[j * 3 + 1];
    float dz = sR[i * 3 + 2] - sR[j * 3 + 2];
    float x = 1.0f / sqrtf(dx * dx + dy * dy + dz * dz);
    xs[m * DESC + p] = x;
    local = x * x;
  }
  for (int off = 16; off > 0; off >>= 1) local += __shfl_xor(local, off, 32);
  if ((tid & 31) == 0) sSum[tid >> 5] = local;
  __syncthreads();
  if (tid == 0) {
    float s = 0.0f;
    for (int w = 0; w < 8; ++w) s += sSum[w];
    xs_n2[m] = kQ2() * s;
  }
}

// ---------------------------------------------------------------------------
// k1: per-training-row scalars. s_t = q^2*||xst_t||^2 ; c_t = q*(xst_t . Jx_t)
// One wave per row; 1500 blocks * 8 waves = 12000 rows exactly.
// ---------------------------------------------------------------------------
__global__ void k1_train(const float* __restrict__ xst, const float* __restrict__ Jx,
                         float* __restrict__ s_t, float* __restrict__ c_t) {
  const int row  = (blockIdx.x * blockDim.x + threadIdx.x) >> 5;
  const int lane = threadIdx.x & 31;
  const float* xr = xst + row * DESC;
  const float* jr = Jx  + row * DESC;
  float s = 0.0f, c = 0.0f;
  for (int d = lane; d < DESC; d += 32) {
    float x = xr[d];
    s += x * x;
    c += x * jr[d];
  }
  for (int off = 16; off > 0; off >>= 1) {
    s += __shfl_xor(s, off, 32);
    c += __shfl_xor(c, off, 32);
  }
  if (lane == 0) { s_t[row] = kQ2() * s; c_t[row] = kQ() * c; }
}

// ---------------------------------------------------------------------------
// k2: WMMA pass 1. G1 = xs@xst^T, G2 = xs@Jx^T (M=64, N=12000, K=210), then the
// Matern-5/2 elementwise math on the accumulator tile. Writes w1,w2 [M,T] and
// deterministic per-(m, t-tile) partials of sum(w1) and sum(w2*dot).
// Grid: 375 blocks x 256 threads. Wave w: mtile = w>>1, t-tile = blk*2+(w&1).
// ---------------------------------------------------------------------------
__global__ void k2_pass1(const float* __restrict__ xs,  const float* __restrict__ xst,
                         const float* __restrict__ Jx,  const float* __restrict__ xs_n2,
                         const float* __restrict__ s_t, const float* __restrict__ c_t,
                         float* __restrict__ W1, float* __restrict__ W2,
                         float* __restrict__ P1, float* __restrict__ P2) {
  const int wave = threadIdx.x >> 5;
  const int lane = threadIdx.x & 31;
  const int half = lane >> 4;          // K-split half of the wave (ISA 7.12.2)
  const int lr   = lane & 15;
  const int m0   = (wave >> 1) * 16;
  const int pt   = blockIdx.x * 2 + (wave & 1);   // t-tile index in [0,750)
  const int t0   = pt * 16;
  const int t    = t0 + lr;
  const int mrow = m0 + lr;

  const float* xsrow = xs  + mrow * DESC;   // A rows (per-lane row m)
  const float* xtrow = xst + t * DESC;      // B cols (per-lane col t), contiguous in k
  const float* jrow  = Jx  + t * DESC;

  v8f acc1 = {}; v8f acc2 = {};
  int k0 = 0;
#pragma unroll 4
  for (; k0 + 4 <= DESC; k0 += 4) {         // 52 full K-steps
    const int ka = k0 + 2 * half;
    v2f a, b1, b2;
    a.x  = xsrow[ka]; a.y  = xsrow[ka + 1];
    b1.x = xtrow[ka]; b1.y = xtrow[ka + 1];
    b2.x = jrow[ka];  b2.y = jrow[ka + 1];
    acc1 = __builtin_amdgcn_wmma_f32_16x16x4_f32(false, a, false, b1, (short)0, acc1, false, false);
    acc2 = __builtin_amdgcn_wmma_f32_16x16x4_f32(false, a, false, b2, (short)0, acc2, false, false);
  }
  { // tail: K = 208..209 valid, 210..211 masked to zero (half==1 lanes)
    const int ka = k0 + 2 * half;
    v2f a = {0.0f, 0.0f}, b1 = {0.0f, 0.0f}, b2 = {0.0f, 0.0f};
    if (ka < DESC)     { a.x = xsrow[ka];     b1.x = xtrow[ka];     b2.x = jrow[ka];     }
    if (ka + 1 < DESC) { a.y = xsrow[ka + 1]; b1.y = xtrow[ka + 1]; b2.y = jrow[ka + 1]; }
    acc1 = __builtin_amdgcn_wmma_f32_16x16x4_f32(false, a, false, b1, (short)0, acc1, false, false);
    acc2 = __builtin_amdgcn_wmma_f32_16x16x4_f32(false, a, false, b2, (short)0, acc2, false, false);
  }

  const float st = s_t[t];
  const float ct = c_t[t];
#pragma unroll
  for (int g = 0; g < 8; ++g) {
    const int m = m0 + g + 8 * half;          // C/D layout: vgpr g -> row g (+8 upper half)
    const float g1 = acc1[g];
    const float g2 = acc2[g];
    float d2  = xs_n2[m] + st - 2.0f * kQ2() * g1;   // ||q xs - q xst||^2
    float xd  = sqrtf(fmaxf(d2, 0.0f));
    float dot = kQ() * g2 - ct;                       // x_diffs . Jx
    float e   = kK0() * __expf(-xd);
    float w1  = e * dot;
    float w2  = e * (1.0f + xd);
    W1[m * TROWS + t] = w1;
    W2[m * TROWS + t] = w2;
    // deterministic 16-lane tree reduction over this tile's t range
    float r1 = w1, r2 = w2 * dot;
    for (int off = 8; off > 0; off >>= 1) {
      r1 += __shfl_xor(r1, off, 16);
      r2 += __shfl_xor(r2, off, 16);
    }
    if (lr == 0) {
      P1[m * TTILES + pt] = r1;   // each (m,pt) written exactly once grid-wide
      P2[m * TTILES + pt] = r2;
    }
  }
}

// ---------------------------------------------------------------------------
// k3: WMMA pass 2. A1 = w1@xst, A2 = w2@Jx  (M=64, N=210, K=12000).
// 56 output tiles (4 m-tiles x 14 d-tiles) = 7 blocks x 8 waves; full K per
// wave -> no cross-block accumulation, fully deterministic.
// ---------------------------------------------------------------------------
__global__ void k3_pass2(const float* __restrict__ W1, const float* __restrict__ W2,
                         const float* __restrict__ xst, const float* __restrict__ Jx,
                         float* __restrict__ A1, float* __restrict__ A2) {
  const int wg   = (blockIdx.x * blockDim.x + threadIdx.x) >> 5;  // 0..55
  const int lane = threadIdx.x & 31;
  const int half = lane >> 4;
  const int lr   = lane & 15;
  const int m0   = (wg / 14) * 16;
  const int d0   = (wg % 14) * 16;
  const int mrow = m0 + lr;
  const int d    = d0 + lr;
  const bool dv  = (d < DESC);
  const int dc   = dv ? d : 0;                 // clamped column for safe loads

  const float* w1row = W1 + mrow * TROWS;
  const float* w2row = W2 + mrow * TROWS;

  v8f acc1 = {}; v8f acc2 = {};
#pragma unroll 2
  for (int k0 = 0; k0 < TROWS; k0 += 4) {      // 3000 steps, TROWS % 4 == 0
    const int ka = k0 + 2 * half;
    v2f a1, a2, b1, b2;
    a1.x = w1row[ka]; a1.y = w1row[ka + 1];
    a2.x = w2row[ka]; a2.y = w2row[ka + 1];
    float bx1 = xst[ka * DESC + dc], by1 = xst[(ka + 1) * DESC + dc];
    float bx2 = Jx [ka * DESC + dc], by2 = Jx [(ka + 1) * DESC + dc];
    b1.x = dv ? bx1 : 0.0f; b1.y = dv ? by1 : 0.0f;
    b2.x = dv ? bx2 : 0.0f; b2.y = dv ? by2 : 0.0f;
    acc1 = __builtin_amdgcn_wmma_f32_16x16x4_f32(false, a1, false, b1, (short)0, acc1, false, false);
    acc2 = __builtin_amdgcn_wmma_f32_16x16x4_f32(false, a2, false, b2, (short)0, acc2, false, false);
  }
#pragma unroll
  for (int g = 0; g < 8; ++g) {
    const int m = m0 + g + 8 * half;
    if (dv) {
      A1[m * DESC + d] = acc1[g];
      A2[m * DESC + d] = acc2[g];
    }
  }
}

// ---------------------------------------------------------------------------
// k4: reduce partials (deterministic), build Fs_x, scatter into symmetric NxN
// and contract with diffs/dists^3 -> forces; emit energies.
// out[0..63] = Es ; out[64 + m*63 + b*3 + k] = Fs[m,b,k]
// ---------------------------------------------------------------------------
__global__ void k4_final(const float* __restrict__ Rs, const float* __restrict__ xs,
                         const float* __restrict__ A1, const float* __restrict__ A2,
                         const float* __restrict__ P1, const float* __restrict__ P2,
                         float* __restrict__ out) {
  __shared__ float sR[NATOMS * 3];
  __shared__ float sFx[DESC];
  __shared__ float sW[8], sE[8];
  const int m = blockIdx.x, tid = threadIdx.x;

  // deterministic reduction of 750 partials per m
  float r1 = 0.0f, r2 = 0.0f;
  for (int j = tid; j < TTILES; j += 256) {
    r1 += P1[m * TTILES + j];
    r2 += P2[m * TTILES + j];
  }
  for (int off = 16; off > 0; off >>= 1) {
    r1 += __shfl_xor(r1, off, 32);
    r2 += __shfl_xor(r2, off, 32);
  }
  if ((tid & 31) == 0) { sW[tid >> 5] = r1; sE[tid >> 5] = r2; }
  if (tid < NATOMS * 3) sR[tid] = Rs[m * NATOMS * 3 + tid];
  __syncthreads();
  float w1sum = 0.0f, esacc = 0.0f;
  for (int w = 0; w < 8; ++w) { w1sum += sW[w]; esacc += sE[w]; }

  if (tid < DESC) {
    // Fs_x = (F1 - F2)*STD ; F1 = q*xs*sum(w1) - q*(w1@xst) ; F2 = w2@Jx
    sFx[tid] = kQ() * xs[m * DESC + tid] * w1sum
             - kQ() * A1[m * DESC + tid]
             - A2[m * DESC + tid];
  }
  __syncthreads();

  if (tid < NATOMS * 3) {
    const int b = tid / 3, k = tid % 3;
    const float rbx = sR[b * 3 + 0], rby = sR[b * 3 + 1], rbz = sR[b * 3 + 2];
    const float rbk = sR[b * 3 + k];
    float acc = 0.0f;
    for (int a = 0; a < NATOMS; ++a) {
      if (a == b) continue;
      float dx = sR[a * 3 + 0] - rbx;
      float dy = sR[a * 3 + 1] - rby;
      float dz = sR[a * 3 + 2] - rbz;
      float d2 = dx * dx + dy * dy + dz * dz;
      float dist = sqrtf(d2);
      int i = a > b ? a : b, j = a > b ? b : a;
      int p = (i * (i - 1)) / 2 + j;
      acc += sFx[p] * (1.0f / (d2 * dist)) * (sR[a * 3 + k] - rbk);
    }
    out[MBATCH + m * (NATOMS * 3) + tid] = acc;   // STD = 1
  }
  if (tid == 0) out[m] = esacc * (1.0f / kQ());    // C = 0, STD = 1
}

// ---------------------------------------------------------------------------
extern "C" void kernel_launch(void* const* d_in, const int* in_sizes, int n_in,
                              void* d_out, int out_size, void* d_ws, size_t ws_size,
                              hipStream_t stream) {
  const float* Rs  = (const float*)d_in[0];   // [64,21,3]
  const float* xst = (const float*)d_in[1];   // [12000,210]
  const float* Jx  = (const float*)d_in[2];   // [12000,210]
  float* out = (float*)d_out;                 // 64 (Es) + 64*63 (Fs)

  // workspace carve-up (floats); total ~6.8 MB
  float* ws    = (float*)d_ws;
  float* W1    = ws;                              // 64*12000
  float* W2    = W1 + MBATCH * TROWS;             // 64*12000
  float* s_t   = W2 + MBATCH * TROWS;             // 12000
  float* c_t   = s_t + TROWS;                     // 12000
  float* xsb   = c_t + TROWS;                     // 64*210
  float* xs_n2 = xsb + MBATCH * DESC;             // 64
  float* A1    = xs_n2 + MBATCH;                  // 64*210
  float* A2    = A1 + MBATCH * DESC;              // 64*210
  float* P1    = A2 + MBATCH * DESC;              // 64*750
  float* P2    = P1 + MBATCH * TTILES;            // 64*750

  k0_descr<<<MBATCH, 256, 0, stream>>>(Rs, xsb, xs_n2);
  k1_train<<<TROWS / 8, 256, 0, stream>>>(xst, Jx, s_t, c_t);
  k2_pass1<<<TROWS / 32, 256, 0, stream>>>(xsb, xst, Jx, xs_n2, s_t, c_t, W1, W2, P1, P2);
  k3_pass2<<<7, 256, 0, stream>>>(W1, W2, xst, Jx, A1, A2);
  k4_final<<<MBATCH, 256, 0, stream>>>(Rs, xsb, A1, A2, P1, P2, out);
}